// MultiHeadAttention_39565238731398
// MI455X (gfx1250) — compile-verified
//
#include <hip/hip_runtime.h>
#include <hip/hip_bf16.h>
#include <math.h>

typedef __attribute__((ext_vector_type(2))) float v2f;
typedef __attribute__((ext_vector_type(8))) float v8f;

#define B_SZ   4
#define TGT    1024
#define SRC    1024
#define DMODEL 1024
#define NHEAD  16
#define DK     64
#define MAXLEN 512
#define BAND   1040   // covers u in [0,1038], multiple of 16

__device__ __forceinline__ v8f wmma_f32_k4(v2f a, v2f b, v8f c) {
  // V_WMMA_F32_16X16X4_F32 : D = A(16x4,f32) * B(4x16,f32) + C(16x16,f32)
  return __builtin_amdgcn_wmma_f32_16x16x4_f32(false, a, false, b, (short)0, c,
                                               false, false);
}

__device__ __forceinline__ int iclamp(int x, int lo, int hi) {
  return x < lo ? lo : (x > hi ? hi : x);
}

// ---------------------------------------------------------------------------
// C[M,N] = X[M,K] @ W[N,K]^T + bias[N]    (torch Linear), fp32 WMMA 16x16x4.
// block = 256 threads = 8 waves; each wave owns one 16x16 output tile.
// grid = (N/128, M/16)
// ---------------------------------------------------------------------------
__global__ __launch_bounds__(256) void gemm_bias_wmma(
    const float* __restrict__ X, const float* __restrict__ W,
    const float* __restrict__ bias, float* __restrict__ C,
    int M, int N, int K) {
  const int wave = threadIdx.x >> 5;
  const int lane = threadIdx.x & 31;
  const int half = lane >> 4;   // 0: K pair {0,1}, 1: K pair {2,3}
  const int l    = lane & 15;
  const int m0 = blockIdx.y << 4;
  const int n0 = (blockIdx.x << 7) + (wave << 4);
  if (m0 >= M || n0 >= N) return;

  const float* xr = X + (size_t)(m0 + l) * K;  // A row for this lane
  const float* wr = W + (size_t)(n0 + l) * K;  // B col (= W row) for this lane
  v8f c = {};
  for (int k0 = 0; k0 < K; k0 += 4) {
    const int ka = k0 + (half << 1);
    v2f a; a.x = xr[ka]; a.y = xr[ka + 1];
    v2f b; b.x = wr[ka]; b.y = wr[ka + 1];
    c = wmma_f32_k4(a, b, c);
  }
  const int col = n0 + l;
  const float bv = bias[col];
  const int rbase = m0 + (half << 3);
  for (int r = 0; r < 8; ++r)
    C[(size_t)(rbase + r) * N + col] = c[r] + bv;
}

// ---------------------------------------------------------------------------
// Fused attention per (b, h, 16-row q-tile). block = 128 threads = 4 waves.
// Dynamic LDS: Srow[16*1024] (scores->attn) + Trow[16*BAND] (rel band / Gq).
// ---------------------------------------------------------------------------
__global__ __launch_bounds__(128) void attn_fused(
    const float* __restrict__ Qb, const float* __restrict__ Kb,
    const float* __restrict__ Vb, const float* __restrict__ relk,
    const float* __restrict__ relv, float* __restrict__ attn_out,
    float* __restrict__ ctx) {
  extern __shared__ float smem[];
  float* Srow = smem;              // 16 * 1024
  float* Trow = smem + 16 * 1024;  // 16 * BAND

  const int qt = blockIdx.x, h = blockIdx.y, b = blockIdx.z;
  const int q0 = qt << 4;
  const int tid = threadIdx.x;
  const int wave = tid >> 5, lane = tid & 31;
  const int half = lane >> 4, l = lane & 15;

  // Q/K/V stored as [b, s, h, d] == row-major [B*L, DMODEL]
  const float* Qbh = Qb + (size_t)b * TGT * DMODEL + h * DK;
  const float* Kbh = Kb + (size_t)b * SRC * DMODEL + h * DK;
  const float* Vbh = Vb + (size_t)b * SRC * DMODEL + h * DK;

  // ---- load the 16x64 Q tile as 16 A-fragments (kept in VGPRs) ----
  v2f qa[16];
  {
    const float* qrow = Qbh + (size_t)(q0 + l) * DMODEL;
    for (int f = 0; f < 16; ++f) {
      const int d = (f << 2) + (half << 1);
      qa[f].x = qrow[d];
      qa[f].y = qrow[d + 1];
    }
  }

  // ---- phase 1: Trow[qi][u] = Q[q0+qi,:] . rel_k[clamp(q0-511+u)] ----
  for (int nt = wave; nt < BAND / 16; nt += 4) {
    const int u0 = nt << 4;
    const int p = iclamp(q0 - 511 + u0 + l, 0, 2 * MAXLEN);
    const float* rk = relk + (size_t)p * DK;
    v8f c = {};
    for (int f = 0; f < 16; ++f) {
      const int d = (f << 2) + (half << 1);
      v2f bfr; bfr.x = rk[d]; bfr.y = rk[d + 1];
      c = wmma_f32_k4(qa[f], bfr, c);
    }
    for (int r = 0; r < 8; ++r)
      Trow[(r + (half << 3)) * BAND + u0 + l] = c[r];
  }
  __syncthreads();

  // ---- phase 2: Srow = scale * (Q K^T + band gather) ----
  const float scale = 0.125f;  // 1/sqrt(64)
  for (int kt = wave; kt < SRC / 16; kt += 4) {
    const float* krow = Kbh + (size_t)(kt * 16 + l) * DMODEL;
    v8f c = {};
    for (int f = 0; f < 16; ++f) {
      const int d = (f << 2) + (half << 1);
      v2f bfr; bfr.x = krow[d]; bfr.y = krow[d + 1];
      c = wmma_f32_k4(qa[f], bfr, c);
    }
    for (int r = 0; r < 8; ++r) {
      const int mi = r + (half << 3);
      const int col = kt * 16 + l;
      const int u = mi - col + 1023;  // in [0,1038]
      Srow[mi * 1024 + col] = scale * (c[r] + Trow[mi * BAND + u]);
    }
  }
  __syncthreads();

  // ---- phase 3: softmax per row (8 lanes per row), emit attn weights ----
  {
    const int row = tid >> 3, sub = tid & 7;
    float* rp = Srow + row * 1024;
    float mx = -3.4e38f;
    for (int cc = sub; cc < 1024; cc += 8) mx = fmaxf(mx, rp[cc]);
    for (int off = 4; off > 0; off >>= 1) mx = fmaxf(mx, __shfl_xor(mx, off));
    float sum = 0.f;
    for (int cc = sub; cc < 1024; cc += 8) {
      const float e = __expf(rp[cc] - mx);
      rp[cc] = e;
      sum += e;
    }
    for (int off = 4; off > 0; off >>= 1) sum += __shfl_xor(sum, off);
    const float inv = 1.0f / sum;
    float* ao = attn_out +
                (((size_t)b * NHEAD + h) * TGT + (q0 + row)) * (size_t)SRC;
    for (int cc = sub; cc < 1024; cc += 8) {
      const float a = rp[cc] * inv;
      rp[cc] = a;
      ao[cc] = a;
    }
  }
  __syncthreads();

  // ---- phase 4a: fold attn row into banded Gq (deterministic gather) ----
  {
    int lo = 511 - q0;  if (lo < 0) lo = 0;
    int hi = 1535 - q0; if (hi > 1038) hi = 1038;
    for (int idx = tid; idx < 16 * BAND; idx += 128) {
      const int qi = idx / BAND;
      const int u = idx - qi * BAND;
      float acc = 0.f;
      if (u >= lo && u <= hi) {
        const int k1 = qi + 1023 - u;  // unclamped preimage
        if (u == lo) {
          for (int k = (k1 < 0 ? 0 : k1); k < 1024; ++k)
            acc += Srow[qi * 1024 + k];
        } else if (u == hi) {
          const int ke = (k1 > 1023 ? 1023 : k1);
          for (int k = 0; k <= ke; ++k) acc += Srow[qi * 1024 + k];
        } else if (k1 >= 0 && k1 < 1024) {
          acc = Srow[qi * 1024 + k1];
        }
      }
      Trow[idx] = acc;  // Trow now holds Gq
    }
  }
  __syncthreads();

  // ---- phase 4b: ctx tile = attn @ V  +  Gq @ rel_v[band] ----
  {
    const int n0 = wave << 4;  // each wave: 16 of the 64 d_k columns
    v8f c = {};
    for (int k0 = 0; k0 < 1024; k0 += 4) {
      const int ka = k0 + (half << 1);
      v2f a; a.x = Srow[l * 1024 + ka]; a.y = Srow[l * 1024 + ka + 1];
      v2f bfr;
      bfr.x = Vbh[(size_t)ka * DMODEL + n0 + l];
      bfr.y = Vbh[(size_t)(ka + 1) * DMODEL + n0 + l];
      c = wmma_f32_k4(a, bfr, c);
    }
    for (int u0 = 0; u0 < BAND; u0 += 4) {
      const int ua = u0 + (half << 1);
      v2f a; a.x = Trow[l * BAND + ua]; a.y = Trow[l * BAND + ua + 1];
      const int p0 = iclamp(q0 - 511 + ua, 0, 2 * MAXLEN);
      const int p1 = iclamp(q0 - 511 + ua + 1, 0, 2 * MAXLEN);
      v2f bfr;
      bfr.x = relv[(size_t)p0 * DK + n0 + l];
      bfr.y = relv[(size_t)p1 * DK + n0 + l];
      c = wmma_f32_k4(a, bfr, c);
    }
    for (int r = 0; r < 8; ++r) {
      const int q = q0 + r + (half << 3);
      ctx[((size_t)b * TGT + q) * DMODEL + h * DK + n0 + l] = c[r];
    }
  }
}

// ---------------------------------------------------------------------------
extern "C" void kernel_launch(void* const* d_in, const int* in_sizes, int n_in,
                              void* d_out, int out_size, void* d_ws,
                              size_t ws_size, hipStream_t stream) {
  (void)in_sizes; (void)n_in; (void)out_size; (void)ws_size;

  const float* query = (const float*)d_in[0];
  const float* key   = (const float*)d_in[1];
  const float* value = (const float*)d_in[2];
  const float* W_q   = (const float*)d_in[3];
  const float* b_q   = (const float*)d_in[4];
  const float* W_k   = (const float*)d_in[5];
  const float* b_k   = (const float*)d_in[6];
  const float* W_v   = (const float*)d_in[7];
  const float* b_v   = (const float*)d_in[8];
  const float* W_o   = (const float*)d_in[9];
  const float* b_o   = (const float*)d_in[10];
  const float* rel_k = (const float*)d_in[11];
  const float* rel_v = (const float*)d_in[12];

  float* out  = (float*)d_out;                       // [4,1024,1024]
  float* attn = out + (size_t)B_SZ * TGT * DMODEL;   // [4,16,1024,1024]

  const size_t actN = (size_t)B_SZ * TGT * DMODEL;   // 4 Mi floats
  float* Qb  = (float*)d_ws;
  float* Kb  = Qb + actN;
  float* Vb  = Kb + actN;
  float* ctx = Vb + actN;                            // 64 MB workspace total

  const int M = B_SZ * TGT;                          // 4096
  dim3 gg(DMODEL / 128, M / 16);                     // (8, 256)
  gemm_bias_wmma<<<gg, 256, 0, stream>>>(query, W_q, b_q, Qb, M, DMODEL, DMODEL);
  gemm_bias_wmma<<<gg, 256, 0, stream>>>(key,   W_k, b_k, Kb, M, DMODEL, DMODEL);
  gemm_bias_wmma<<<gg, 256, 0, stream>>>(value, W_v, b_v, Vb, M, DMODEL, DMODEL);

  const size_t smem_bytes = (size_t)(16 * 1024 + 16 * BAND) * sizeof(float);
  attn_fused<<<dim3(TGT / 16, NHEAD, B_SZ), 128, smem_bytes, stream>>>(
      Qb, Kb, Vb, rel_k, rel_v, attn, ctx);

  gemm_bias_wmma<<<gg, 256, 0, stream>>>(ctx, W_o, b_o, out, M, DMODEL, DMODEL);
}